// AdaptiveNoiseCanceller_46918222742096
// MI455X (gfx1250) — compile-verified
//
#include <hip/hip_runtime.h>
#include <hip/hip_bf16.h>
#include <stdint.h>

// LMS adaptive noise canceller for MI455X (gfx1250, wave32).
// B=32 batches, C=128 channels -> 4096 independent serial recurrences.
// 1 thread per (b,c); weights + windows in registers; inputs staged into LDS
// with the gfx1250 async-copy engine (ASYNCcnt), double-buffered.
//
// Recurrence is restructured with one step of lookahead:
//   y_n = w_{n-1}.x_n = p_n + g_{n-1} * s_n,
//   p_n = w_{n-2}.x_n           (one full iteration of slack -> off chain)
//   s_n = x_{n-1}.x_n           (no weight dependence; sliding product window)
// so the loop-carried chain is only fma -> sub -> mul (3 VALU ops) instead of
// update -> dot-tree -> sub -> mul (~8), moving the kernel from latency-bound
// to issue-bound on a dedicated SIMD per chain.

#define FO 10
#define B_ 32
#define L_ 8192
#define C_ 128
#define TS 32                 // timesteps per staged chunk
#define NCHUNK (L_ / TS)      // 256
#define CHUNK_ELEMS (TS * C_) // 4096 floats = 16 KB per array per buffer

// Issue 16 async b128 global->LDS copies per wave for chunk t into buffer p.
// s_ref buffer layout: slot j holds ref[b, t*TS + j - 1, :]  (shifted by one
// timestep so step n reads ref[n-1] at j = n - t*TS). For t==0 the j==0 row
// is never consumed; its lanes are clamped to a safe in-bounds address.
__device__ __forceinline__ void copy_chunk(const float* __restrict__ refg,
                                           const float* __restrict__ noisyg,
                                           uint32_t lref, uint32_t lnoi,
                                           int b, int t, int p, int tid) {
  const size_t e0 = ((size_t)b * L_ + (size_t)t * TS) * C_;
  const uint64_t nbase = (uint64_t)(uintptr_t)(noisyg + e0);
  const uint64_t rbase = (uint64_t)(uintptr_t)(refg + e0) - (uint64_t)(C_ * 4);
  const uint32_t lr = lref + (uint32_t)(p * CHUNK_ELEMS * 4);
  const uint32_t ln = lnoi + (uint32_t)(p * CHUNK_ELEMS * 4);
#pragma unroll
  for (int k = 0; k < 8; ++k) {
    const uint32_t off = 16u * (uint32_t)(k * C_ + tid);     // bytes, 16B/lane
    const uint32_t roff = (t == 0 && off < 512u) ? (off + 512u) : off;
    asm volatile("global_load_async_to_lds_b128 %0, %1, %2"
                 :: "v"(ln + off), "v"(off), "s"(nbase) : "memory");
    asm volatile("global_load_async_to_lds_b128 %0, %1, %2"
                 :: "v"(lr + roff), "v"(roff), "s"(rbase) : "memory");
  }
}

// Fully-unrolled processing of one 32-timestep chunk. J0=FO only for chunk 0.
// State: w  = w_{n-2} at loop top, g1 = 2*MU*e_{n-1},
//        xw[k] = ref[n-2-k] (k=0..10) before the shift,
//        pm[k] = ref[n-2-k]*ref[n-3-k] (terms of s_{n-1}),
//        s = s_{n-1}.
template <int J0>
__device__ __forceinline__ void process_chunk(const float* __restrict__ sref,
                                              const float* __restrict__ snoi,
                                              float* __restrict__ out,
                                              int b, int t, int c,
                                              float (&w)[FO], float (&xw)[FO + 1],
                                              float (&pm)[FO], float& s, float& g1) {
  float* outp = out + ((size_t)b * L_ + (size_t)t * TS) * C_ + c;
  // Exact recompute of s from the product window once per chunk: caps the
  // fp drift of the incremental update at ~32 steps' worth of rounding.
  s = (((pm[0] + pm[1]) + (pm[2] + pm[3])) + ((pm[4] + pm[5]) + (pm[6] + pm[7]))) +
      (pm[8] + pm[9]);
#pragma unroll
  for (int j = J0; j < TS; ++j) {
    const float xn = sref[j * C_ + c];   // = ref[b, n-1, c]
    const float d  = snoi[j * C_ + c];   // = noisy[b, n, c]

    // s_n = s_{n-1} + ref[n-1]*ref[n-2] - ref[n-11]*ref[n-12]
    const float pmn = xn * xw[0];
    s = (s + pmn) - pm[FO - 1];
#pragma unroll
    for (int k = FO - 1; k > 0; --k) pm[k] = pm[k - 1];   // renamed, no moves
    pm[0] = pmn;
#pragma unroll
    for (int k = FO; k > 0; --k) xw[k] = xw[k - 1];
    xw[0] = xn;                           // xw[k] = ref[n-1-k], k=0..10

    // p_n = w_{n-2} . x_n : two 5-deep FMA chains (depth covered by the
    // one-iteration slack on w), 11 instructions total.
    const float pa = fmaf(w[8], xw[8], fmaf(w[6], xw[6],
                       fmaf(w[4], xw[4], fmaf(w[2], xw[2], w[0] * xw[0]))));
    const float pb = fmaf(w[9], xw[9], fmaf(w[7], xw[7],
                       fmaf(w[5], xw[5], fmaf(w[3], xw[3], w[1] * xw[1]))));
    const float p = pa + pb;

    // Loop-carried chain: fma -> sub -> mul.
    const float y = fmaf(g1, s, p);       // = w_{n-1} . x_n  (exact algebra)
    const float e = d - y;
    outp[(size_t)j * C_] = e;

    // w_{n-1} = w_{n-2} + g_{n-1} * x_{n-1}: 10 independent FMAs, off chain.
#pragma unroll
    for (int k = 0; k < FO; ++k) w[k] = fmaf(g1, xw[k + 1], w[k]);

    g1 = 0.02f * e;                       // 2 * MU * e_n
  }
}

__global__ __launch_bounds__(C_, 1)
void lms_anc_kernel(const float* __restrict__ noisy,
                    const float* __restrict__ refsig,
                    const float* __restrict__ wts,
                    float* __restrict__ out) {
  __shared__ float s_ref[2 * CHUNK_ELEMS];    // 32 KB
  __shared__ float s_noi[2 * CHUNK_ELEMS];    // 32 KB
  const int tid = threadIdx.x;
  const int c = tid;
  const int b = blockIdx.x;

  // Raw LDS byte offsets (generic LDS address low 32 bits == DS offset).
  const uint32_t lref = (uint32_t)(uintptr_t)(void*)s_ref;
  const uint32_t lnoi = (uint32_t)(uintptr_t)(void*)s_noi;

  float w[FO], xw[FO + 1], pm[FO];
  float s = 0.0f;
  float g1 = 0.0f;  // g_{n-1}=0 for the first step: makes w_{n-2}==w_init and
                    // kills the (undefined) s/xw[10] contributions exactly.
#pragma unroll
  for (int k = 0; k < FO; ++k) {
    w[k] = wts[((size_t)b * FO + k) * C_ + c];
    // window before first step (n=10): xw[k] = ref[b, 9-k, c]
    xw[k] = refsig[((size_t)b * L_ + (FO - 1 - k)) * C_ + c];
    // denoised[:, :FO, :] = noisy[:, :FO, :]
    out[((size_t)b * L_ + k) * C_ + c] = noisy[((size_t)b * L_ + k) * C_ + c];
  }
  xw[FO] = 0.0f;  // stands in for ref[-1]; only ever multiplied by g1==0
#pragma unroll
  for (int k = 0; k < FO; ++k) pm[k] = xw[k] * xw[k + 1];

  // Prologue: stage chunks 0 and 1, wait for chunk 0 (in-order completion).
  copy_chunk(refsig, noisy, lref, lnoi, b, 0, 0, tid);
  copy_chunk(refsig, noisy, lref, lnoi, b, 1, 1, tid);
  asm volatile("s_wait_asynccnt 0x10" ::: "memory");
  __syncthreads();

  for (int t = 0; t < NCHUNK; ++t) {
    const int p = t & 1;
    if (t == 0)
      process_chunk<FO>(s_ref + p * CHUNK_ELEMS, s_noi + p * CHUNK_ELEMS,
                        out, b, t, c, w, xw, pm, s, g1);
    else
      process_chunk<0>(s_ref + p * CHUNK_ELEMS, s_noi + p * CHUNK_ELEMS,
                       out, b, t, c, w, xw, pm, s, g1);
    __syncthreads();                       // everyone done reading buffer p
    if (t + 2 < NCHUNK) {
      copy_chunk(refsig, noisy, lref, lnoi, b, t + 2, p, tid);
      asm volatile("s_wait_asynccnt 0x10" ::: "memory");  // chunk t+1 landed
    } else {
      asm volatile("s_wait_asynccnt 0x0" ::: "memory");   // drain tail
    }
    __syncthreads();                       // buffer for chunk t+1 visible
  }
}

extern "C" void kernel_launch(void* const* d_in, const int* in_sizes, int n_in,
                              void* d_out, int out_size, void* d_ws, size_t ws_size,
                              hipStream_t stream) {
  (void)in_sizes; (void)n_in; (void)out_size; (void)d_ws; (void)ws_size;
  const float* noisy  = (const float*)d_in[0];
  const float* refsig = (const float*)d_in[1];
  const float* wts    = (const float*)d_in[2];
  float* out = (float*)d_out;
  lms_anc_kernel<<<dim3(B_), dim3(C_), 0, stream>>>(noisy, refsig, wts, out);
}